// pLoss_all_fidelity_21517786153440
// MI455X (gfx1250) — compile-verified
//
#include <hip/hip_runtime.h>
#include <math.h>

typedef float v2f __attribute__((ext_vector_type(2)));
typedef float v8f __attribute__((ext_vector_type(8)));

#define NSTATES 96
#define LDIM    12
#define TILES_N 6      // 96/16 state tiles for GEMM1 output
#define KC1     3      // K chunks for GEMM1 (12 = 3*4)
#define KC2     24     // K chunks for GEMM2 (96 = 24*4)
#define JSTRIDE 100    // padded row stride (floats) -> conflict-free banks
#define EPS_F   1e-8f

// ---------------------------------------------------------------------------
// Main kernel: one wave = one 16-row batch tile. 8 waves / block.
// ---------------------------------------------------------------------------
__global__ __launch_bounds__(256) void ploss_main(
    const float* __restrict__ f,      // (B,12) f32
    const int*   __restrict__ y,      // (B,12) int
    const float* __restrict__ S,      // (96,12) f32
    float*       __restrict__ out,    // [0..2]=loss, [3..] = pMargin (B,12)
    float*       __restrict__ ws,     // per-block loss partials (3 per block)
    int Bn)
{
    // B1: GEMM1 B-operand (S^T) in native WMMA B layout: float2 per (n,kc,lane)
    __shared__ float b1[TILES_N * KC1 * 32 * 2];
    // B2: GEMM2 B-operand (mask) in native WMMA B layout: float2 per (kc,lane)
    __shared__ float b2[KC2 * 32 * 2];
    // J staging for the D-layout -> A-layout transpose, per wave 16 x JSTRIDE
    __shared__ float jlds[8 * 16 * JSTRIDE];
    __shared__ float red[8 * 3];

    const int tid  = threadIdx.x;
    const int wave = tid >> 5;
    const int lane = tid & 31;
    const int col  = lane & 15;   // N index within a 16-wide tile
    const int h    = lane >> 4;   // half-wave selector

    // ---- Stage B1: element(k, s) of S^T for state tile n, K chunk kc.
    // B layout (16x4 per chunk): lane = column, reg0 holds K = 4kc+2h, reg1 = +1.
    for (int idx = tid; idx < TILES_N * KC1 * 32; idx += 256) {
        int l  = idx & 31;
        int kc = (idx >> 5) % KC1;
        int n  = idx / (32 * KC1);
        int s  = n * 16 + (l & 15);
        int k  = 4 * kc + 2 * (l >> 4);
        b1[idx * 2 + 0] = S[s * LDIM + k];
        b1[idx * 2 + 1] = S[s * LDIM + k + 1];
    }
    // ---- Stage B2: mask = (S > 0), B layout for GEMM2 (K = state, N = j).
    for (int idx = tid; idx < KC2 * 32; idx += 256) {
        int l  = idx & 31;
        int kc = idx >> 5;
        int j  = l & 15;
        int s  = 4 * kc + 2 * (l >> 4);
        b2[idx * 2 + 0] = (j < LDIM && S[s * LDIM + j] > 0.f) ? 1.f : 0.f;
        b2[idx * 2 + 1] = (j < LDIM && S[(s + 1) * LDIM + j] > 0.f) ? 1.f : 0.f;
    }
    __syncthreads();

    float ls0 = 0.f, ls1 = 0.f, ls2 = 0.f;

    long long r0 = ((long long)blockIdx.x * 8 + wave) * 16;
    if (r0 < Bn) {
        // ================= GEMM1: potential tile (16 rows x 96 states) ======
        v8f c[TILES_N];
        #pragma unroll
        for (int n = 0; n < TILES_N; n++) c[n] = (v8f){0.f,0.f,0.f,0.f,0.f,0.f,0.f,0.f};

        const long long row = r0 + col;           // A layout: lane -> batch row
        const bool rok = (row < (long long)Bn);

        #pragma unroll
        for (int kc = 0; kc < KC1; kc++) {
            v2f a;
            if (rok) {
                const float* fp = f + row * LDIM + 4 * kc + 2 * h;  // 8B aligned
                a[0] = fp[0]; a[1] = fp[1];
            } else {
                a[0] = 0.f; a[1] = 0.f;
            }
            #pragma unroll
            for (int n = 0; n < TILES_N; n++) {
                v2f b;
                int bi = ((n * KC1 + kc) * 32 + lane) * 2;
                b[0] = b1[bi]; b[1] = b1[bi + 1];
                c[n] = __builtin_amdgcn_wmma_f32_16x16x4_f32(
                           false, a, false, b, (short)0, c[n], false, false);
            }
        }

        // ============ softmax stats per row, fully in registers =============
        // D layout: lane holds (m = v + 8h, s = 16n + col). Row stats need a
        // reduce over the 6 groups + 16 lanes of this half-wave (width-16 xor).
        float z[8];
        #pragma unroll
        for (int v = 0; v < 8; v++) {
            float mv = c[0][v];
            #pragma unroll
            for (int n = 1; n < TILES_N; n++) mv = fmaxf(mv, c[n][v]);
            for (int off = 1; off < 16; off <<= 1)
                mv = fmaxf(mv, __shfl_xor(mv, off, 16));
            float sv = 0.f;
            #pragma unroll
            for (int n = 0; n < TILES_N; n++) {
                float e = __expf(c[n][v] - mv);
                c[n][v] = e;                    // J value
                sv += e;
            }
            for (int off = 1; off < 16; off <<= 1)
                sv += __shfl_xor(sv, off, 16);
            z[v] = sv;   // z for row m = v + 8h : exactly this lane's GEMM2 rows
        }

        // ---- D-layout -> LDS (row-major J, padded) for the A-layout reload.
        float* jw = jlds + wave * 16 * JSTRIDE;
        #pragma unroll
        for (int n = 0; n < TILES_N; n++)
            #pragma unroll
            for (int v = 0; v < 8; v++)
                jw[(v + 8 * h) * JSTRIDE + n * 16 + col] = c[n][v];

        // ================= GEMM2: J (16x96) x mask (96x16) ==================
        v8f d = (v8f){0.f,0.f,0.f,0.f,0.f,0.f,0.f,0.f};
        #pragma unroll
        for (int kc = 0; kc < KC2; kc++) {
            v2f a2, b;
            const float* jp = jw + col * JSTRIDE + 4 * kc + 2 * h;  // 8B aligned
            a2[0] = jp[0]; a2[1] = jp[1];
            int bi = (kc * 32 + lane) * 2;
            b[0] = b2[bi]; b[1] = b2[bi + 1];
            d = __builtin_amdgcn_wmma_f32_16x16x4_f32(
                    false, a2, false, b, (short)0, d, false, false);
        }

        // ============ normalize, store pMargin, loss contributions ==========
        #pragma unroll
        for (int v = 0; v < 8; v++) {
            long long rr = r0 + v + 8 * h;
            int j = col;
            if (rr < (long long)Bn && j < LDIM) {
                float p = d[v] / z[v];
                out[3 + rr * LDIM + j] = p;
                float g = (float)y[rr * LDIM + j];
                float t = 1.f - (sqrtf(p * g + EPS_F) +
                                 sqrtf((1.f - p) * (1.f - g) + EPS_F));
                if (j == 0)      ls0 += t;
                else if (j < 6)  ls1 += t;
                else             ls2 += t;
            }
        }
    }

    // ---- deterministic block reduction of the three loss partials ----------
    for (int off = 16; off > 0; off >>= 1) {
        ls0 += __shfl_xor(ls0, off, 32);
        ls1 += __shfl_xor(ls1, off, 32);
        ls2 += __shfl_xor(ls2, off, 32);
    }
    if (lane == 0) {
        red[wave * 3 + 0] = ls0;
        red[wave * 3 + 1] = ls1;
        red[wave * 3 + 2] = ls2;
    }
    __syncthreads();
    if (tid == 0) {
        float a = 0.f, b = 0.f, cc = 0.f;
        for (int w = 0; w < 8; w++) {
            a  += red[w * 3 + 0];
            b  += red[w * 3 + 1];
            cc += red[w * 3 + 2];
        }
        ws[(long long)blockIdx.x * 3 + 0] = a;
        ws[(long long)blockIdx.x * 3 + 1] = b;
        ws[(long long)blockIdx.x * 3 + 2] = cc;
    }
}

// ---------------------------------------------------------------------------
// Final deterministic reduction of per-block partials into the 3 losses.
// ---------------------------------------------------------------------------
__global__ __launch_bounds__(256) void ploss_reduce(
    const float* __restrict__ ws, float* __restrict__ out, int nblk, int Bn)
{
    __shared__ float r[8 * 3];
    int tid = threadIdx.x;
    float a = 0.f, b = 0.f, c = 0.f;
    for (int i = tid; i < nblk; i += 256) {
        a += ws[(long long)i * 3 + 0];
        b += ws[(long long)i * 3 + 1];
        c += ws[(long long)i * 3 + 2];
    }
    for (int off = 16; off > 0; off >>= 1) {
        a += __shfl_xor(a, off, 32);
        b += __shfl_xor(b, off, 32);
        c += __shfl_xor(c, off, 32);
    }
    if ((tid & 31) == 0) {
        int w = tid >> 5;
        r[w * 3 + 0] = a; r[w * 3 + 1] = b; r[w * 3 + 2] = c;
    }
    __syncthreads();
    if (tid == 0) {
        float A = 0.f, B = 0.f, C = 0.f;
        for (int w = 0; w < 8; w++) {
            A += r[w * 3 + 0]; B += r[w * 3 + 1]; C += r[w * 3 + 2];
        }
        out[0] = A / (float)Bn;
        out[1] = B / ((float)Bn * 5.f);
        out[2] = C / ((float)Bn * 6.f);
    }
}

extern "C" void kernel_launch(void* const* d_in, const int* in_sizes, int n_in,
                              void* d_out, int out_size, void* d_ws, size_t ws_size,
                              hipStream_t stream)
{
    const float* f = (const float*)d_in[0];
    const int*   y = (const int*)d_in[1];
    const float* S = (const float*)d_in[2];
    float* out = (float*)d_out;
    float* ws  = (float*)d_ws;

    int Bn   = in_sizes[0] / LDIM;            // batch size
    int nblk = (Bn + 127) / 128;              // 8 waves x 16 rows per block

    ploss_main<<<nblk, 256, 0, stream>>>(f, y, S, out, ws, Bn);
    ploss_reduce<<<1, 256, 0, stream>>>(ws, out, nblk, Bn);
}